// GATODEFunc_6897717477531
// MI455X (gfx1250) — compile-verified
//
#include <hip/hip_runtime.h>
#include <hip/hip_bf16.h>

#define NNODES 50000
#define NEDGES 800000
#define HID    64
#define HEADS  4
#define NEG_SLOPE 0.2f
#define LN_EPS 1e-5f
#define OUTW  (HEADS * HID)   // 256

typedef float v2f __attribute__((ext_vector_type(2)));
typedef float v8f __attribute__((ext_vector_type(8)));

// ---------- ordered-uint encoding for float atomicMax ----------
__device__ __forceinline__ unsigned f2ord(float f) {
    unsigned u = __float_as_uint(f);
    return (u & 0x80000000u) ? ~u : (u | 0x80000000u);
}
__device__ __forceinline__ float ord2f(unsigned u) {
    return (u & 0x80000000u) ? __uint_as_float(u & 0x7fffffffu)
                             : __uint_as_float(~u);
}

// ---------- pre-reduce attention vectors ----------
// vsrc[k*4+h] = sum_c W[k,h*64+c]*att_src[h,c]; same vdst; medge[k*4+h] from W_edge/att_edge
__global__ void precompute_kernel(const float* __restrict__ W,
                                  const float* __restrict__ attsrc,
                                  const float* __restrict__ attdst,
                                  const float* __restrict__ Wedge,
                                  const float* __restrict__ attedge,
                                  float* __restrict__ vsrc,
                                  float* __restrict__ vdst,
                                  float* __restrict__ medge) {
    int t  = threadIdx.x;          // 256 threads
    int k  = t >> 2;
    int hh = t & 3;
    float ss = 0.f, dd = 0.f;
    for (int c = 0; c < HID; ++c) {
        float w = W[k * OUTW + hh * HID + c];
        ss += w * attsrc[hh * HID + c];
        dd += w * attdst[hh * HID + c];
    }
    vsrc[k * 4 + hh] = ss;
    vdst[k * 4 + hh] = dd;
    if (t < 12) {                  // k in 0..2, h in 0..3
        int ke = t >> 2, he = t & 3;
        float m = 0.f;
        for (int c = 0; c < HID; ++c)
            m += Wedge[ke * OUTW + he * HID + c] * attedge[he * HID + c];
        medge[ke * 4 + he] = m;
    }
}

// ---------- x = h @ W via V_WMMA_F32_16X16X4_F32 ----------
// one wave per 16x16 tile of x[N,256]; K=64 in 16 steps of 4
__global__ void gemm_x_kernel(const float* __restrict__ h,
                              const float* __restrict__ W,
                              float* __restrict__ x) {
    const int ntn   = OUTW / 16;                 // 16 tiles along N'
    int wave = blockIdx.x * (blockDim.x >> 5) + (threadIdx.x >> 5);
    int lane = threadIdx.x & 31;
    int tile_m = wave / ntn;
    int tile_n = wave % ntn;
    if (tile_m * 16 >= NNODES) return;           // wave-uniform: EXEC stays all-ones

    int half = lane >> 4;                        // 0 or 1 (K-split)
    int l16  = lane & 15;
    int row  = tile_m * 16 + l16;                // A-matrix row (M)
    int rowc = (row < NNODES) ? row : (NNODES - 1);
    int col  = tile_n * 16 + l16;                // B-matrix col (N')

    v8f c = {};
    #pragma unroll
    for (int k0 = 0; k0 < HID; k0 += 4) {
        int ka = k0 + half * 2;
        v2f a, b;
        a.x = h[(size_t)rowc * HID + ka];
        a.y = h[(size_t)rowc * HID + ka + 1];
        b.x = W[(size_t)(ka)     * OUTW + col];
        b.y = W[(size_t)(ka + 1) * OUTW + col];
        c = __builtin_amdgcn_wmma_f32_16x16x4_f32(false, a, false, b,
                                                  (short)0, c, false, false);
    }
    #pragma unroll
    for (int r = 0; r < 8; ++r) {
        int orow = tile_m * 16 + r + half * 8;   // C layout: lanes>=16 hold M=r+8
        if (orow < NNODES) x[(size_t)orow * OUTW + col] = c[r];
    }
}

// ---------- a_src/a_dst = h @ vsrc/vdst ----------
__global__ void node_attn_kernel(const float* __restrict__ h,
                                 const float* __restrict__ vsrc,
                                 const float* __restrict__ vdst,
                                 float* __restrict__ asrc,
                                 float* __restrict__ adst) {
    __shared__ float sv[HID * 4], sd[HID * 4];
    for (int i = threadIdx.x; i < HID * 4; i += blockDim.x) {
        sv[i] = vsrc[i]; sd[i] = vdst[i];
    }
    __syncthreads();
    int n = blockIdx.x * blockDim.x + threadIdx.x;
    if (n >= NNODES) return;
    float s0 = 0, s1 = 0, s2 = 0, s3 = 0, d0 = 0, d1 = 0, d2 = 0, d3 = 0;
    for (int k = 0; k < HID; ++k) {
        float hv = h[(size_t)n * HID + k];
        s0 += hv * sv[k * 4 + 0]; s1 += hv * sv[k * 4 + 1];
        s2 += hv * sv[k * 4 + 2]; s3 += hv * sv[k * 4 + 3];
        d0 += hv * sd[k * 4 + 0]; d1 += hv * sd[k * 4 + 1];
        d2 += hv * sd[k * 4 + 2]; d3 += hv * sd[k * 4 + 3];
    }
    asrc[n * 4 + 0] = s0; asrc[n * 4 + 1] = s1; asrc[n * 4 + 2] = s2; asrc[n * 4 + 3] = s3;
    adst[n * 4 + 0] = d0; adst[n * 4 + 1] = d1; adst[n * 4 + 2] = d2; adst[n * 4 + 3] = d3;
}

// ---------- edge pass 1: logits, leaky-relu, segment max ----------
__global__ void edge_pass1_kernel(const int* __restrict__ ei,
                                  const float* __restrict__ eattr,
                                  const float* __restrict__ asrc,
                                  const float* __restrict__ adst,
                                  const float* __restrict__ medge,
                                  float* __restrict__ ealpha,
                                  unsigned* __restrict__ amax) {
    int e = blockIdx.x * blockDim.x + threadIdx.x;
    if (e >= NEDGES) return;
    int s = ei[e];
    int d = ei[NEDGES + e];
    float e0 = eattr[(size_t)e * 3 + 0];
    float e1 = eattr[(size_t)e * 3 + 1];
    float e2 = eattr[(size_t)e * 3 + 2];
    #pragma unroll
    for (int hh = 0; hh < HEADS; ++hh) {
        float ae = e0 * medge[0 * 4 + hh] + e1 * medge[1 * 4 + hh] + e2 * medge[2 * 4 + hh];
        float a  = asrc[s * 4 + hh] + adst[d * 4 + hh] + ae;
        a = (a > 0.f) ? a : NEG_SLOPE * a;
        ealpha[(size_t)e * 4 + hh] = a;
        atomicMax(&amax[d * 4 + hh], f2ord(a));
    }
}

// ---------- edge pass 2: exp + segment sum ----------
__global__ void edge_pass2_kernel(const int* __restrict__ ei,
                                  float* __restrict__ ealpha,
                                  const unsigned* __restrict__ amax,
                                  float* __restrict__ esum) {
    int e = blockIdx.x * blockDim.x + threadIdx.x;
    if (e >= NEDGES) return;
    int d = ei[NEDGES + e];
    #pragma unroll
    for (int hh = 0; hh < HEADS; ++hh) {
        float a  = ealpha[(size_t)e * 4 + hh];
        float m  = ord2f(amax[d * 4 + hh]);
        float ex = __expf(a - m);
        ealpha[(size_t)e * 4 + hh] = ex;
        atomicAdd(&esum[d * 4 + hh], ex);
    }
}

// ---------- edge pass 3: weighted scatter-add, heads fused ----------
// 64 threads per edge; acc[n,c] += sum_h x[src,h,c] * coef[e,h]
__global__ void edge_pass3_kernel(const int* __restrict__ ei,
                                  const float* __restrict__ x,
                                  const float* __restrict__ ealpha,
                                  const float* __restrict__ esum,
                                  float* __restrict__ acc) {
    long long idx = (long long)blockIdx.x * blockDim.x + threadIdx.x;
    int e = (int)(idx >> 6);
    int c = (int)(idx & 63);
    if (e >= NEDGES) return;
    int s = ei[e];
    int d = ei[NEDGES + e];
    const float* xr = x + (size_t)s * OUTW;
    __builtin_prefetch(xr + c, 0, 1);            // -> global_prefetch_b8 (L2-resident gather)
    float coef[HEADS];
    #pragma unroll
    for (int hh = 0; hh < HEADS; ++hh)
        coef[hh] = ealpha[(size_t)e * 4 + hh] / (esum[d * 4 + hh] + 1e-16f);
    float v = xr[0 * HID + c] * coef[0] + xr[1 * HID + c] * coef[1]
            + xr[2 * HID + c] * coef[2] + xr[3 * HID + c] * coef[3];
    atomicAdd(&acc[(size_t)d * HID + c], v);
}

// ---------- node finalize: head mean + bias + LayerNorm + SiLU ----------
__global__ void node_final_kernel(const float* __restrict__ acc,
                                  const float* __restrict__ bias,
                                  const float* __restrict__ gamma,
                                  const float* __restrict__ beta,
                                  float* __restrict__ out) {
    int n = blockIdx.x;
    int c = threadIdx.x;                         // 64 threads
    __shared__ float red[HID];
    float v = acc[(size_t)n * HID + c] * (1.0f / HEADS) + bias[c];
    red[c] = v;
    __syncthreads();
    for (int off = 32; off > 0; off >>= 1) {
        if (c < off) red[c] += red[c + off];
        __syncthreads();
    }
    float mu = red[0] * (1.0f / HID);
    __syncthreads();
    float dv = v - mu;
    red[c] = dv * dv;
    __syncthreads();
    for (int off = 32; off > 0; off >>= 1) {
        if (c < off) red[c] += red[c + off];
        __syncthreads();
    }
    float var = red[0] * (1.0f / HID);
    float y = dv * rsqrtf(var + LN_EPS) * gamma[c] + beta[c];
    out[(size_t)n * HID + c] = y / (1.0f + __expf(-y));   // y * sigmoid(y)
}

extern "C" void kernel_launch(void* const* d_in, const int* in_sizes, int n_in,
                              void* d_out, int out_size, void* d_ws, size_t ws_size,
                              hipStream_t stream) {
    // inputs in setup_inputs() order
    const float* h       = (const float*)d_in[1];
    const int*   ei      = (const int*)  d_in[2];   // [2,E]
    const float* eattr   = (const float*)d_in[3];   // [E,3]
    const float* W       = (const float*)d_in[4];   // [64,256]
    const float* attsrc  = (const float*)d_in[5];   // [1,4,64]
    const float* attdst  = (const float*)d_in[6];
    const float* Wedge   = (const float*)d_in[7];   // [3,256]
    const float* attedge = (const float*)d_in[8];
    const float* bias    = (const float*)d_in[9];
    const float* gamma   = (const float*)d_in[10];
    const float* beta    = (const float*)d_in[11];
    float* out = (float*)d_out;

    // workspace carve-up
    float*    x      = (float*)d_ws;                       // N*256
    float*    asrc   = x      + (size_t)NNODES * OUTW;     // N*4
    float*    adst   = asrc   + (size_t)NNODES * 4;        // N*4
    float*    ealpha = adst   + (size_t)NNODES * 4;        // E*4
    unsigned* amax   = (unsigned*)(ealpha + (size_t)NEDGES * 4); // N*4
    float*    esum   = (float*)(amax + (size_t)NNODES * 4);      // N*4
    float*    acc    = esum   + (size_t)NNODES * 4;        // N*64
    float*    vsrc   = acc    + (size_t)NNODES * HID;      // 256
    float*    vdst   = vsrc + HID * 4;                     // 256
    float*    medge  = vdst + HID * 4;                     // 12

    // zero amax (ordered -inf == 0), esum, acc (contiguous region)
    size_t zero_bytes = ((size_t)NNODES * 4 + (size_t)NNODES * 4 + (size_t)NNODES * HID) * sizeof(float);
    hipMemsetAsync((void*)amax, 0, zero_bytes, stream);

    precompute_kernel<<<1, 256, 0, stream>>>(W, attsrc, attdst, Wedge, attedge, vsrc, vdst, medge);

    {   // GEMM: one wave per 16x16 tile, 8 waves per block
        int tiles_m = (NNODES + 15) / 16;                  // 3125
        int waves   = tiles_m * (OUTW / 16);               // 50000
        int blocks  = (waves + 7) / 8;
        gemm_x_kernel<<<blocks, 256, 0, stream>>>(h, W, x);
    }

    node_attn_kernel<<<(NNODES + 255) / 256, 256, 0, stream>>>(h, vsrc, vdst, asrc, adst);

    edge_pass1_kernel<<<(NEDGES + 255) / 256, 256, 0, stream>>>(ei, eattr, asrc, adst, medge, ealpha, amax);
    edge_pass2_kernel<<<(NEDGES + 255) / 256, 256, 0, stream>>>(ei, ealpha, amax, esum);

    {   // 64 threads per edge
        long long thr = (long long)NEDGES * HID;
        int blocks = (int)((thr + 255) / 256);
        edge_pass3_kernel<<<blocks, 256, 0, stream>>>(ei, x, ealpha, esum, acc);
    }

    node_final_kernel<<<NNODES, HID, 0, stream>>>(acc, bias, gamma, beta, out);
}